// SemanticModule_79474074845471
// MI455X (gfx1250) — compile-verified
//
#include <hip/hip_runtime.h>
#include <hip/hip_bf16.h>
#include <float.h>

// ---------------------------------------------------------------------------
// Hetero-GNN for MI455X (gfx1250, wave32).
//   * scatter/aggregate via L2 atomics (feature tables fit in 192MB L2)
//   * all dense GEMMs via V_WMMA_F32_16X16X4_F32 (fp32-exact)
//   * weights pre-packed (padded + k-pair interleaved) and staged in LDS
// ---------------------------------------------------------------------------

typedef __attribute__((ext_vector_type(2))) float v2f;
typedef __attribute__((ext_vector_type(8))) float v8f;

#define NS_N 300000
#define NB_N 150000
#define HID 32

// packed weight block: 3 matrices (16 kp x 32 n x 2) + 32 bias = 3104 floats
#define PACKW_N 3104
// packed output head: 16 kp x 64 n x 2 + 64 bias = 2112 floats
#define PACKO_N 2112

__device__ __forceinline__ v8f wmma4(v2f a, v2f b, v8f c) {
    // D = A(16x4,f32) * B(4x16,f32) + C(16x16,f32)
    return __builtin_amdgcn_wmma_f32_16x16x4_f32(
        false, a, false, b, (short)0, c, false, false);
}

__device__ __forceinline__ void atomicMaxF(float* addr, float val) {
    // float-max via int/uint atomics; consistent with -FLT_MAX init
    if (val >= 0.0f) atomicMax((int*)addr, __float_as_int(val));
    else             atomicMin((unsigned int*)addr, __float_as_uint(val));
}

// ---------------------------------------------------------------------------
__global__ void fill4_kernel(float4* __restrict__ p, long n4, float v) {
    long i = (long)blockIdx.x * blockDim.x + threadIdx.x;
    if (i < n4) p[i] = make_float4(v, v, v, v);
}

// pad input [n,6] -> [n,32] (cols 6..31 = 0)
__global__ void pad_copy_kernel(const float* __restrict__ in,
                                float* __restrict__ out, int n) {
    long i = (long)blockIdx.x * blockDim.x + threadIdx.x;
    if (i >= (long)n * 32) return;
    int node = (int)(i >> 5), c = (int)(i & 31);
    out[i] = (c < 6) ? in[(long)node * 6 + c] : 0.0f;
}

__global__ void count_kernel(const int* __restrict__ dst,
                             float* __restrict__ cnt, int E) {
    int e = blockIdx.x * blockDim.x + threadIdx.x;
    if (e < E) atomicAdd(&cnt[dst[e]], 1.0f);
}

// ---------------------------------------------------------------------------
// Pack per-layer weights: zero-pad rows >= cin, interleave k-pairs so each
// B fragment is one contiguous 8-byte load, pre-sum root weights & biases.
//   out[mat*1024 + (kp*32+n)*2 + lo] = W[2kp+lo][n]   (mat 0:W1 1:W2 2:Wr1+Wr2)
//   out[3072+n] = b1[n]+b2[n]
// ---------------------------------------------------------------------------
__global__ void pack_weights_kernel(const float* __restrict__ W1,
                                    const float* __restrict__ W2,
                                    const float* __restrict__ Wr1,
                                    const float* __restrict__ Wr2,
                                    const float* __restrict__ b1,
                                    const float* __restrict__ b2,
                                    int cin, float* __restrict__ out) {
    int i = blockIdx.x * blockDim.x + threadIdx.x;
    if (i >= PACKW_N) return;
    if (i < 3072) {
        int mat = i >> 10;
        int r   = i & 1023;
        int kp  = r >> 6;
        int nn  = (r >> 1) & 31;
        int lo  = r & 1;
        int k   = kp * 2 + lo;
        float v = 0.0f;
        if (k < cin) {
            if (mat == 0)      v = W1[k * 32 + nn];
            else if (mat == 1) v = W2[k * 32 + nn];
            else               v = Wr1[k * 32 + nn] + Wr2[k * 32 + nn];
        }
        out[i] = v;
    } else {
        int nn = i - 3072;
        out[i] = b1[nn] + b2[nn];
    }
}

// pack output head: out[(kp*64+n)*2+lo] = W[2kp+lo][n]; out[2048+n] = b[n]
__global__ void pack_out_kernel(const float* __restrict__ W,
                                const float* __restrict__ b,
                                float* __restrict__ out) {
    int i = blockIdx.x * blockDim.x + threadIdx.x;
    if (i >= PACKO_N) return;
    if (i < 2048) {
        int kp = i >> 7;
        int nn = (i >> 1) & 63;
        int lo = i & 1;
        out[i] = W[(kp * 2 + lo) * 64 + nn];
    } else {
        out[i] = b[i - 2048];
    }
}

// ---------------------------------------------------------------------------
// scatter-add / scatter-max: `chunks` threads per edge, 4 cols each
// ---------------------------------------------------------------------------
__global__ void scatter_add_kernel(const float* __restrict__ x,
                                   const int* __restrict__ src,
                                   const int* __restrict__ dst,
                                   float* __restrict__ agg,
                                   int E, int cin, int chunks) {
    long tid = (long)blockIdx.x * blockDim.x + threadIdx.x;
    long e = tid / chunks;
    if (e >= E) return;
    int c0 = (int)(tid % chunks) * 4;
    long s = src[e], d = dst[e];
    const float* xr = x + s * 32 + c0;
    float* ar = agg + d * 32 + c0;
    int m = cin - c0;
    if (m >= 4) {
        float4 v = *(const float4*)xr;
        atomicAdd(ar + 0, v.x);
        atomicAdd(ar + 1, v.y);
        atomicAdd(ar + 2, v.z);
        atomicAdd(ar + 3, v.w);
    } else {
        for (int i = 0; i < m; ++i) atomicAdd(ar + i, xr[i]);
    }
}

__global__ void scatter_max_kernel(const float* __restrict__ x,
                                   const int* __restrict__ src,
                                   const int* __restrict__ dst,
                                   float* __restrict__ agg,
                                   int E, int cin, int chunks) {
    long tid = (long)blockIdx.x * blockDim.x + threadIdx.x;
    long e = tid / chunks;
    if (e >= E) return;
    int c0 = (int)(tid % chunks) * 4;
    long s = src[e], d = dst[e];
    const float* xr = x + s * 32 + c0;
    float* ar = agg + d * 32 + c0;
    int m = cin - c0;
    if (m >= 4) {
        float4 v = *(const float4*)xr;
        atomicMaxF(ar + 0, v.x);
        atomicMaxF(ar + 1, v.y);
        atomicMaxF(ar + 2, v.z);
        atomicMaxF(ar + 3, v.w);
    } else {
        for (int i = 0; i < m; ++i) atomicMaxF(ar + i, xr[i]);
    }
}

// ---------------------------------------------------------------------------
// Combine (in-place): x[0:32] = (residual? x:0)
//                   + agg1*W1 + agg2(max-fixed)*W2 + x*Wroot_sum + bias_sum
// agg1 optionally divided by max(cnt,1) (mean). One wave owns 16 rows.
// Packed weights staged in LDS; fully-unrolled branch-free WMMA chain.
// ---------------------------------------------------------------------------
template <int KSTEPS>
__global__ __launch_bounds__(256)
void combine_kernel(float* __restrict__ x,
                    const float* __restrict__ agg1,
                    const float* __restrict__ agg2,
                    const float* __restrict__ cnt,
                    const float* __restrict__ pack,
                    int n, int residual, int useCnt) {
    __shared__ float lw[PACKW_N];
    for (int i = threadIdx.x; i < PACKW_N; i += 256) lw[i] = pack[i];
    __syncthreads();

    int wave = (int)((blockIdx.x * blockDim.x + threadIdx.x) >> 5);
    int lane = threadIdx.x & 31;
    int base = wave * 16;
    if (base >= n) return;                 // wave-uniform: EXEC stays all-1s
    int lm = lane & 15;
    int hi = lane >> 4;
    long row = base + lm;                  // A-fragment row for this lane
    float inv1 = useCnt ? (1.0f / fmaxf(cnt[row], 1.0f)) : 1.0f;

    v8f acc0 = {}, acc1 = {};

    #pragma unroll
    for (int s = 0; s < 3; ++s) {
        const float* A  = (s == 0) ? agg1 : (s == 1) ? agg2 : x;
        const float* Wl = lw + s * 1024;
        #pragma unroll
        for (int ks = 0; ks < KSTEPS; ++ks) {
            int kk = ks * 4 + hi * 2;      // this lane's K pair (even -> 8B aligned)
            int kp = ks * 2 + hi;
            v2f a = *(const v2f*)(A + row * 32 + kk);
            if (s == 0 && useCnt) { a[0] *= inv1; a[1] *= inv1; }
            if (s == 1) {                  // max-agg: empty segment -> 0
                a[0] = (a[0] == -FLT_MAX) ? 0.0f : a[0];
                a[1] = (a[1] == -FLT_MAX) ? 0.0f : a[1];
            }
            v2f b0 = *(const v2f*)(Wl + (kp * 32 + lm) * 2);
            v2f b1 = *(const v2f*)(Wl + (kp * 32 + 16 + lm) * 2);
            acc0 = wmma4(a, b0, acc0);     // cols 0..15
            acc1 = wmma4(a, b1, acc1);     // cols 16..31
        }
    }

    float bias0 = lw[3072 + lm];
    float bias1 = lw[3072 + 16 + lm];
    #pragma unroll
    for (int r = 0; r < 8; ++r) {
        long orow = base + r + hi * 8;     // C/D layout: m = r + 8*hi
        float o0 = acc0[r] + bias0;
        float o1 = acc1[r] + bias1;
        if (residual) {
            o0 += x[orow * 32 + lm];
            o1 += x[orow * 32 + 16 + lm];
        }
        x[orow * 32 + lm] = o0;
        x[orow * 32 + 16 + lm] = o1;
    }
}

// ---------------------------------------------------------------------------
// Output head: out[n,64] = relu(xb[n,32]) @ W[32,64] + b[64]
// ---------------------------------------------------------------------------
__global__ __launch_bounds__(256)
void out_kernel(const float* __restrict__ xb,
                const float* __restrict__ pack,
                float* __restrict__ out, int n) {
    __shared__ float lw[PACKO_N];
    for (int i = threadIdx.x; i < PACKO_N; i += 256) lw[i] = pack[i];
    __syncthreads();

    int wave = (int)((blockIdx.x * blockDim.x + threadIdx.x) >> 5);
    int lane = threadIdx.x & 31;
    int base = wave * 16;
    if (base >= n) return;
    int lm = lane & 15, hi = lane >> 4;
    long row = base + lm;
    v8f acc[4] = {v8f{}, v8f{}, v8f{}, v8f{}};
    #pragma unroll
    for (int ks = 0; ks < 8; ++ks) {
        int kk = ks * 4 + hi * 2;
        int kp = ks * 2 + hi;
        v2f a = *(const v2f*)(xb + row * 32 + kk);
        a[0] = fmaxf(a[0], 0.0f);
        a[1] = fmaxf(a[1], 0.0f);
        #pragma unroll
        for (int t = 0; t < 4; ++t) {
            v2f b = *(const v2f*)(lw + (kp * 64 + t * 16 + lm) * 2);
            acc[t] = wmma4(a, b, acc[t]);
        }
    }
    #pragma unroll
    for (int t = 0; t < 4; ++t) {
        float bias = lw[2048 + t * 16 + lm];
        #pragma unroll
        for (int r = 0; r < 8; ++r) {
            long orow = base + r + hi * 8;
            out[orow * 64 + t * 16 + lm] = acc[t][r] + bias;
        }
    }
}

// ---------------------------------------------------------------------------
extern "C" void kernel_launch(void* const* d_in, const int* in_sizes, int n_in,
                              void* d_out, int out_size, void* d_ws, size_t ws_size,
                              hipStream_t stream) {
    (void)in_sizes; (void)n_in; (void)out_size; (void)ws_size;

    // ---- decode inputs (setup_inputs insertion order) ----
    const float* xs_in = (const float*)d_in[0];   // [NS,6]
    const float* xb_in = (const float*)d_in[1];   // [NB,6]
    int p = 2;
    // params: 5 layers x rels[tp,rb,bc,sc] x {Wrel, brel, Wroot}
    const float *Wrel[5][4], *brel[5][4], *Wroot[5][4];
    for (int L = 0; L < 5; ++L)
        for (int r = 0; r < 4; ++r) {
            Wrel[L][r]  = (const float*)d_in[p++];
            brel[L][r]  = (const float*)d_in[p++];
            Wroot[L][r] = (const float*)d_in[p++];
        }
    const float* linW = (const float*)d_in[p++];  // [32,64]
    const float* linb = (const float*)d_in[p++];  // [64]
    const int* ei[4];                             // tp, rb, bc, sc (each [2,E])
    for (int r = 0; r < 4; ++r) ei[r] = (const int*)d_in[p++];
    const int E[4] = {1200000, 600000, 600000, 1200000};

    // ---- workspace layout (fp32) ----
    float* ws = (float*)d_ws;
    float* xs    = ws; ws += (size_t)NS_N * 32;   // stroke features
    float* xb    = ws; ws += (size_t)NB_N * 32;   // brep features
    float* aggS1 = ws; ws += (size_t)NS_N * 32;   // tp (add)
    float* aggS2 = ws; ws += (size_t)NS_N * 32;   // sc (max)
    float* aggB1 = ws; ws += (size_t)NB_N * 32;   // rb (mean-sum)
    float* aggB2 = ws; ws += (size_t)NB_N * 32;   // bc (max)
    float* cnt   = ws; ws += (size_t)NB_N;        // rb in-degree
    float* packS = ws; ws += PACKW_N;             // packed stroke-combine weights
    float* packB = ws; ws += PACKW_N;             // packed brep-combine weights
    float* packO = ws; ws += PACKO_N;             // packed output head

    auto cdiv = [](long a, long b) { return (unsigned)((a + b - 1) / b); };
    const int BLK = 256;

    // head embed + mean counts (counts are layer-invariant)
    pad_copy_kernel<<<cdiv((long)NS_N * 32, BLK), BLK, 0, stream>>>(xs_in, xs, NS_N);
    pad_copy_kernel<<<cdiv((long)NB_N * 32, BLK), BLK, 0, stream>>>(xb_in, xb, NB_N);
    fill4_kernel<<<cdiv(NB_N / 4, BLK), BLK, 0, stream>>>((float4*)cnt, NB_N / 4, 0.0f);
    count_kernel<<<cdiv(E[1], BLK), BLK, 0, stream>>>(ei[1] + E[1], cnt, E[1]);

    for (int L = 0; L < 5; ++L) {
        int cin = (L == 0) ? 6 : HID;
        int chunks = (cin + 3) / 4;

        fill4_kernel<<<cdiv((long)NS_N * 8, BLK), BLK, 0, stream>>>((float4*)aggS1, (long)NS_N * 8, 0.0f);
        fill4_kernel<<<cdiv((long)NS_N * 8, BLK), BLK, 0, stream>>>((float4*)aggS2, (long)NS_N * 8, -FLT_MAX);
        fill4_kernel<<<cdiv((long)NB_N * 8, BLK), BLK, 0, stream>>>((float4*)aggB1, (long)NB_N * 8, 0.0f);
        fill4_kernel<<<cdiv((long)NB_N * 8, BLK), BLK, 0, stream>>>((float4*)aggB2, (long)NB_N * 8, -FLT_MAX);

        // pack weights for this layer (stroke: tp+sc, brep: rb+bc)
        pack_weights_kernel<<<cdiv(PACKW_N, BLK), BLK, 0, stream>>>(
            Wrel[L][0], Wrel[L][3], Wroot[L][0], Wroot[L][3],
            brel[L][0], brel[L][3], cin, packS);
        pack_weights_kernel<<<cdiv(PACKW_N, BLK), BLK, 0, stream>>>(
            Wrel[L][1], Wrel[L][2], Wroot[L][1], Wroot[L][2],
            brel[L][1], brel[L][2], cin, packB);

        // tp: stroke->stroke (add)
        scatter_add_kernel<<<cdiv((long)E[0] * chunks, BLK), BLK, 0, stream>>>(
            xs, ei[0], ei[0] + E[0], aggS1, E[0], cin, chunks);
        // rb: stroke->brep (mean numerator)
        scatter_add_kernel<<<cdiv((long)E[1] * chunks, BLK), BLK, 0, stream>>>(
            xs, ei[1], ei[1] + E[1], aggB1, E[1], cin, chunks);
        // bc: brep->brep (max)
        scatter_max_kernel<<<cdiv((long)E[2] * chunks, BLK), BLK, 0, stream>>>(
            xb, ei[2], ei[2] + E[2], aggB2, E[2], cin, chunks);
        // sc: stroke->stroke (max)
        scatter_max_kernel<<<cdiv((long)E[3] * chunks, BLK), BLK, 0, stream>>>(
            xs, ei[3], ei[3] + E[3], aggS2, E[3], cin, chunks);

        // combine (in-place): stroke then brep
        if (L == 0) {
            combine_kernel<2><<<cdiv((long)(NS_N / 16) * 32, BLK), BLK, 0, stream>>>(
                xs, aggS1, aggS2, cnt, packS, NS_N, 0, 0);
            combine_kernel<2><<<cdiv((long)(NB_N / 16) * 32, BLK), BLK, 0, stream>>>(
                xb, aggB1, aggB2, cnt, packB, NB_N, 0, 1);
        } else {
            combine_kernel<8><<<cdiv((long)(NS_N / 16) * 32, BLK), BLK, 0, stream>>>(
                xs, aggS1, aggS2, cnt, packS, NS_N, 1, 0);
            combine_kernel<8><<<cdiv((long)(NB_N / 16) * 32, BLK), BLK, 0, stream>>>(
                xb, aggB1, aggB2, cnt, packB, NB_N, 1, 1);
        }
    }

    // final head: relu(xb) @ linW + linb -> d_out [NB,64]
    pack_out_kernel<<<cdiv(PACKO_N, BLK), BLK, 0, stream>>>(linW, linb, packO);
    out_kernel<<<cdiv((long)(NB_N / 16) * 32, BLK), BLK, 0, stream>>>(
        xb, packO, (float*)d_out, NB_N);
}